// CND4_996432413159
// MI455X (gfx1250) — compile-verified
//
#include <hip/hip_runtime.h>
#include <math.h>

// CN-D4 coordination counting:
//   rc      = rc_table[species]                         [N]
//   rcij    = rc[src] + rc[dst]                         [E]
//   rij     = dist / BOHR
//   CNij    = 0.5*(1 + erf(-K0*(rij/rcij - 1))) * switch
//   CNi     = segment_sum(CNij by src)                  [N]
// d_out = [ CNi (N floats) | CNij (E floats) ]
//
// Memory-bound: ~272 MB streamed @ 23.3 TB/s. No matrix structure -> no WMMA.
// Strategy: L2-resident rc/CNi, NT 128-bit streaming, no-return f32 atomics.

#define K0F       7.5f
#define INV_BOHR  1.8897261254578281f   // 1 / 0.52917721067121

typedef __attribute__((ext_vector_type(4))) float v4f;
typedef __attribute__((ext_vector_type(4))) int   v4i;

// ---- Kernel A: per-node covalent radius gather + zero the CNi accumulator.
__global__ void cn_node_prep(const int* __restrict__ species,
                             const float* __restrict__ rc_table,
                             float* __restrict__ rc_node,
                             float* __restrict__ CNi,
                             int n_nodes) {
  int i = blockIdx.x * blockDim.x + threadIdx.x;
  if (i < n_nodes) {
    rc_node[i] = rc_table[species[i]];
    CNi[i]     = 0.0f;
  }
}

// Zero-only variant (fallback path when workspace is unavailable).
__global__ void cn_zero(float* __restrict__ CNi, int n_nodes) {
  int i = blockIdx.x * blockDim.x + threadIdx.x;
  if (i < n_nodes) CNi[i] = 0.0f;
}

__device__ __forceinline__ float cn_edge_math(float rcs, float rcd,
                                              float dist, float w) {
  float rij = dist * INV_BOHR;
  // -K0*(rij/rcij - 1) == K0 - K0*rij/rcij
  float x  = K0F - __fdividef(K0F * rij, rcs + rcd);
  return 0.5f * (1.0f + erff(x)) * w;
}

// ---- Kernel B: main edge kernel. 4 edges/thread via 128-bit NT loads,
// persistent grid-stride with prefetch of the next stride.
__global__ void cn_edge_kernel(const int* __restrict__ edge_src,
                               const int* __restrict__ edge_dst,
                               const float* __restrict__ dist,
                               const float* __restrict__ sw,
                               const float* __restrict__ rc_node,
                               float* __restrict__ CNi,
                               float* __restrict__ CNij,
                               int n_edges) {
  const int nvec = n_edges >> 2;               // groups of 4 edges
  const int gsz  = gridDim.x * blockDim.x;
  const int gid  = blockIdx.x * blockDim.x + threadIdx.x;

  for (int g = gid; g < nvec; g += gsz) {
    const int e  = g << 2;
    const int gn = g + gsz;
    if (gn < nvec) {                           // hide latency of next stride
      const int en = gn << 2;
      __builtin_prefetch(&edge_src[en], 0, 0); // -> global_prefetch_b8
      __builtin_prefetch(&edge_dst[en], 0, 0);
      __builtin_prefetch(&dist[en],     0, 0);
      __builtin_prefetch(&sw[en],       0, 0);
    }
    v4i s = __builtin_nontemporal_load((const v4i*)(edge_src + e));
    v4i d = __builtin_nontemporal_load((const v4i*)(edge_dst + e));
    v4f r = __builtin_nontemporal_load((const v4f*)(dist + e));
    v4f w = __builtin_nontemporal_load((const v4f*)(sw + e));
    v4f out;
#pragma unroll
    for (int k = 0; k < 4; ++k) {
      float cn = cn_edge_math(rc_node[s[k]], rc_node[d[k]], r[k], w[k]);
      out[k] = cn;
      atomicAdd(&CNi[s[k]], cn);               // no-return global_atomic_add_f32
    }
    __builtin_nontemporal_store(out, (v4f*)(CNij + e));
  }

  // Tail: n_edges % 4 leftover edges (<= 3), scalar.
  const int tail_base = nvec << 2;
  const int idx = tail_base + gid;
  if (idx < n_edges) {
    int sN = edge_src[idx], dN = edge_dst[idx];
    float cn = cn_edge_math(rc_node[sN], rc_node[dN], dist[idx], sw[idx]);
    CNij[idx] = cn;
    atomicAdd(&CNi[sN], cn);
  }
}

// ---- Fallback edge kernel: double-indirect gather, no workspace needed.
__global__ void cn_edge_kernel_nows(const int* __restrict__ edge_src,
                                    const int* __restrict__ edge_dst,
                                    const float* __restrict__ dist,
                                    const float* __restrict__ sw,
                                    const int* __restrict__ species,
                                    const float* __restrict__ rc_table,
                                    float* __restrict__ CNi,
                                    float* __restrict__ CNij,
                                    int n_edges) {
  const int gsz = gridDim.x * blockDim.x;
  for (int i = blockIdx.x * blockDim.x + threadIdx.x; i < n_edges; i += gsz) {
    int sN = edge_src[i], dN = edge_dst[i];
    float rcs = rc_table[species[sN]];
    float rcd = rc_table[species[dN]];
    float cn = cn_edge_math(rcs, rcd, dist[i], sw[i]);
    CNij[i] = cn;
    atomicAdd(&CNi[sN], cn);
  }
}

extern "C" void kernel_launch(void* const* d_in, const int* in_sizes, int n_in,
                              void* d_out, int out_size, void* d_ws, size_t ws_size,
                              hipStream_t stream) {
  const int*   species   = (const int*)d_in[0];
  const int*   edge_src  = (const int*)d_in[1];
  const int*   edge_dst  = (const int*)d_in[2];
  const float* distances = (const float*)d_in[3];
  const float* sw        = (const float*)d_in[4];
  const float* rc_table  = (const float*)d_in[5];
  const int n_nodes = in_sizes[0];
  const int n_edges = in_sizes[1];

  float* CNi  = (float*)d_out;            // first n_nodes floats
  float* CNij = (float*)d_out + n_nodes;  // next n_edges floats

  const int tpb = 256;                    // 8 wave32 waves per block
  const int nblk_nodes = (n_nodes + tpb - 1) / tpb;

  const bool have_ws = (ws_size >= (size_t)n_nodes * sizeof(float));

  if (have_ws) {
    float* rc_node = (float*)d_ws;        // 800 KB, L2-resident
    cn_node_prep<<<nblk_nodes, tpb, 0, stream>>>(species, rc_table, rc_node,
                                                 CNi, n_nodes);
    int nvec = n_edges >> 2;
    int eblk = (nvec + tpb - 1) / tpb;
    if (eblk > 4096) eblk = 4096;         // persistent grid-stride
    if (eblk < 1)    eblk = 1;
    cn_edge_kernel<<<eblk, tpb, 0, stream>>>(edge_src, edge_dst, distances, sw,
                                             rc_node, CNi, CNij, n_edges);
  } else {
    cn_zero<<<nblk_nodes, tpb, 0, stream>>>(CNi, n_nodes);
    int eblk = (n_edges + tpb - 1) / tpb;
    if (eblk > 8192) eblk = 8192;
    if (eblk < 1)    eblk = 1;
    cn_edge_kernel_nows<<<eblk, tpb, 0, stream>>>(edge_src, edge_dst, distances,
                                                  sw, species, rc_table,
                                                  CNi, CNij, n_edges);
  }
}